// AdaptiveEmbedding_59871844107157
// MI455X (gfx1250) — compile-verified
//
#include <hip/hip_runtime.h>

typedef __attribute__((ext_vector_type(2))) float v2f;
typedef __attribute__((ext_vector_type(8))) float v8f;

#define D_PROJ    1024
#define EMB_SCALE 32.0f   // sqrt(1024)

// Zero row for masked-out lanes. Uninitialized __device__ globals live in
// device .bss and are zero-filled at module load; this kernel never writes it,
// so inactive lanes' A-fragments read exact zeros with NO exec divergence.
__device__ __align__(16) float g_zero_row[1024];

// One 16x4 (A: gathered embeddings) x 4x16 (B: proj^T slice) f32 WMMA per
// 4-wide K step, accumulating into the wave's 16x16 f32 C tile.
// Both loads are unconditional b64s -> no exec-mask dance in the hot loop.
__device__ __forceinline__ v8f cluster_pass(v8f acc,
                                            const float* __restrict__ erow,
                                            const float* __restrict__ projRow,
                                            int kOff, int d)
{
#pragma unroll 8
    for (int k = 0; k < d; k += 4) {
        v2f a = *(const v2f*)(erow + k + kOff);     // A[M=lane&15][k+kOff .. +1]
        v2f b = *(const v2f*)(projRow + k + kOff);  // B[k+kOff .. +1][N=lane&15]
        acc = __builtin_amdgcn_wmma_f32_16x16x4_f32(
            /*neg_a=*/false, a, /*neg_b=*/false, b,
            /*c_mod=*/(short)0, acc, /*reuse_a=*/false, /*reuse_b=*/false);
    }
    return acc;
}

__global__ __launch_bounds__(256)
void adaptive_embedding_wmma(const int*   __restrict__ inp,
                             const float* __restrict__ emb0, const float* __restrict__ proj0,
                             const float* __restrict__ emb1, const float* __restrict__ proj1,
                             const float* __restrict__ emb2, const float* __restrict__ proj2,
                             const float* __restrict__ emb3, const float* __restrict__ proj3,
                             float* __restrict__ out)
{
    const int lane    = threadIdx.x & 31;
    const int wave    = threadIdx.x >> 5;              // 8 waves/block
    const int tokBase = blockIdx.x * 16;               // 16 tokens per tile (M)
    const int pBase   = (blockIdx.y * 8 + wave) * 16;  // 16 proj dims per wave (N)

    const int row  = lane & 15;      // M row (A) / N column (B,C,D) owned by this lane
    const int hi   = lane >> 4;      // upper half-wave handles K+2..3 / M+8..15
    const int kOff = hi * 2;

    const int tok = inp[tokBase + row];

    // Cluster membership; clamped local indices keep addresses in-bounds, and
    // inactive lanes are redirected to the zero row (contribution == 0).
    const bool c0 = tok < 20000;
    const bool c1 = (tok >= 20000)  && (tok < 40000);
    const bool c2 = (tok >= 40000)  && (tok < 200000);
    const bool c3 = tok >= 200000;
    const int  l0 = min(max(tok,            0),  19999);
    const int  l1 = min(max(tok - 20000,    0),  19999);
    const int  l2 = min(max(tok - 40000,    0), 159999);
    const int  l3 = min(max(tok - 200000,   0),  67734);

    const float* e0 = c0 ? (emb0 + (size_t)l0 * 1024) : g_zero_row;
    const float* e1 = c1 ? (emb1 + (size_t)l1 * 256)  : g_zero_row;
    const float* e2 = c2 ? (emb2 + (size_t)l2 * 64)   : g_zero_row;
    const float* e3 = c3 ? (emb3 + (size_t)l3 * 16)   : g_zero_row;

    // Warm the gather rows into cache (global_prefetch_b8); rows are 4KB max.
    __builtin_prefetch(e0, 0, 0);
    __builtin_prefetch(e1, 0, 0);
    __builtin_prefetch(e2, 0, 0);
    __builtin_prefetch(e3, 0, 0);

    const int pc = pBase + row;      // proj output-dim row this lane streams for B

    v8f acc = {};
    acc = cluster_pass(acc, e0, proj0 + (size_t)pc * 1024, kOff, 1024);
    acc = cluster_pass(acc, e1, proj1 + (size_t)pc * 256,  kOff, 256);
    acc = cluster_pass(acc, e2, proj2 + (size_t)pc * 64,   kOff, 64);
    acc = cluster_pass(acc, e3, proj3 + (size_t)pc * 16,   kOff, 16);

    // C/D layout: VGPR r -> M=r (lanes 0-15) or M=r+8 (lanes 16-31), N=lane&15.
    // Each of the 8 stores writes two contiguous 64B segments (fully coalesced).
    const size_t outBase = (size_t)tokBase * D_PROJ + (size_t)pBase + row;
#pragma unroll
    for (int r = 0; r < 8; ++r) {
        const int m = r + hi * 8;
        out[outBase + (size_t)m * D_PROJ] = acc[r] * EMB_SCALE;
    }
}

extern "C" void kernel_launch(void* const* d_in, const int* in_sizes, int n_in,
                              void* d_out, int out_size, void* d_ws, size_t ws_size,
                              hipStream_t stream) {
    // setup_inputs() dict order: inp, emb0, proj0, emb1, proj1, emb2, proj2, emb3, proj3
    const int*   inp   = (const int*)  d_in[0];
    const float* emb0  = (const float*)d_in[1];
    const float* proj0 = (const float*)d_in[2];
    const float* emb1  = (const float*)d_in[3];
    const float* proj1 = (const float*)d_in[4];
    const float* emb2  = (const float*)d_in[5];
    const float* proj2 = (const float*)d_in[6];
    const float* emb3  = (const float*)d_in[7];
    const float* proj3 = (const float*)d_in[8];
    float* out = (float*)d_out;

    const int nTok = in_sizes[0];               // 4*4096 = 16384, multiple of 16
    dim3 grid(nTok / 16, D_PROJ / (16 * 8));    // (1024, 8)
    adaptive_embedding_wmma<<<grid, 256, 0, stream>>>(
        inp, emb0, proj0, emb1, proj1, emb2, proj2, emb3, proj3, out);
}